// regGCN_83760452206926
// MI455X (gfx1250) — compile-verified
//
#include <hip/hip_runtime.h>

#define N_NODES  131072
#define N_EDGES  8388608
#define N_GRAPHS 64

typedef __attribute__((ext_vector_type(2))) float v2f;
typedef __attribute__((ext_vector_type(8))) float v8f;

// fp32 global atomic add -> global_atomic_add_f32 (agent scope, L2-resident tables)
__device__ __forceinline__ void gatomic_add(float* p, float v) {
    __hip_atomic_fetch_add(p, v, __ATOMIC_RELAXED, __HIP_MEMORY_SCOPE_AGENT);
}

// ---------------------------------------------------------------- init
__global__ void k_init(unsigned* __restrict__ deg, float* __restrict__ gsum,
                       unsigned* __restrict__ gcnt) {
    int i = blockIdx.x * blockDim.x + threadIdx.x;
    if (i < N_NODES) deg[i] = 0u;
    if (i < N_GRAPHS) { gsum[i] = 0.f; gcnt[i] = 0u; }
}

// ---------------------------------------------------------------- degree
// 4 edges per lane via 16B loads; u32 hardware atomics into 512KB table (L2).
__global__ void k_degree(const int* __restrict__ dst, unsigned* __restrict__ deg) {
    int t = blockIdx.x * blockDim.x + threadIdx.x;
    int4 d = ((const int4*)dst)[t];
    atomicAdd(&deg[d.x], 1u);
    atomicAdd(&deg[d.y], 1u);
    atomicAdd(&deg[d.z], 1u);
    atomicAdd(&deg[d.w], 1u);
}

// ---------------------------------------------------------------- node pass 1
// dinv = rsqrt(deg+1)  (self-loop);  p = dinv*x;  seed agg1 with self-loop term p.
__global__ void k_node1(const float* __restrict__ x, const unsigned* __restrict__ deg,
                        float* __restrict__ dinv, float* __restrict__ p,
                        float* __restrict__ agg1) {
    int n = blockIdx.x * blockDim.x + threadIdx.x;
    float dv = __frsqrt_rn((float)(deg[n] + 1u));
    dinv[n] = dv;
    float pv = dv * x[n];
    p[n] = pv;
    agg1[n] = pv;
}

// ---------------------------------------------------------------- edge scatter
// acc[dst] += val[src]; val/acc are separate buffers so gather never races scatter.
__global__ void k_edge_scatter(const int* __restrict__ src, const int* __restrict__ dst,
                               const float* __restrict__ val, float* __restrict__ acc) {
    int t = blockIdx.x * blockDim.x + threadIdx.x;
    int4 s = ((const int4*)src)[t];
    int4 d = ((const int4*)dst)[t];
    gatomic_add(&acc[d.x], val[s.x]);
    gatomic_add(&acc[d.y], val[s.y]);
    gatomic_add(&acc[d.z], val[s.z]);
    gatomic_add(&acc[d.w], val[s.w]);
}

// ---------------------------------------------------------------- node pass 2
// a1 = dinv*agg1; h1 = relu(a1*W1+b1) (16-wide); t = h1.W2; q = dinv*t; seed agg2.
__global__ void k_node2(const float* __restrict__ dinv, const float* __restrict__ agg1,
                        const float* __restrict__ W1, const float* __restrict__ b1,
                        const float* __restrict__ W2,
                        float* __restrict__ q, float* __restrict__ agg2) {
    int n = blockIdx.x * blockDim.x + threadIdx.x;
    float dv = dinv[n];
    float a1 = dv * agg1[n];
    float t = 0.f;
#pragma unroll
    for (int j = 0; j < 16; ++j)
        t += fmaxf(fmaf(a1, W1[j], b1[j]), 0.f) * W2[j];
    float qv = dv * t;
    q[n] = qv;
    agg2[n] = qv;
}

// ---------------------------------------------------------------- MLP (WMMA)
// Per wave: 16 nodes. h2 = relu(dinv*agg2 + b2); v1[j] = relu(h2*Wl1[j]+bl1[j]).
// V2 = V1[16x16] @ Wl2pad[16x16] via 4x V_WMMA_F32_16X16X4_F32 (K=16).
// A layout (ISA 7.12.2, 32-bit A 16x4): lane L<16 holds A[L][K=0],A[L][K=1] in the
// VGPR pair; lane L>=16 holds A[L-16][K=2],A[L-16][K=3]. So lane (hi,ln) needs
// v1 elements j0=4c+2*hi and j0+1 for node ln -> compute them directly from the
// runtime-indexed (cache-hot) Wl1/bl1 loads: no register array, no select chains.
// B rows mirror A's K split; out-of-range columns (ln>=8) are zeroed by a
// multiply mask on an always-in-range load (col = ln&7) -> fully branchless,
// EXEC stays all-1s across the whole WMMA sequence.
// C layout: VGPR r, lanes 0-15 -> node r; lanes 16-31 -> node r+8; N = lane&15.
__global__ void k_mlp(const float* __restrict__ dinv, const float* __restrict__ agg2,
                      const float* __restrict__ b2,
                      const float* __restrict__ Wl1, const float* __restrict__ bl1,
                      const float* __restrict__ Wl2, const float* __restrict__ bl2,
                      const float* __restrict__ Wl3, const float* __restrict__ bl3,
                      float* __restrict__ y) {
    const int lane = threadIdx.x & 31;
    const int wave = threadIdx.x >> 5;
    const int hi   = lane >> 4;       // 0: lanes 0-15, 1: lanes 16-31
    const int ln   = lane & 15;
    const int base = blockIdx.x * 128 + wave * 16;
    const int node = base + ln;

    float h2 = fmaxf(fmaf(dinv[node], agg2[node], 0.f) + b2[0], 0.f);

    const int   k0  = hi ? 2 : 0;          // this lane's K offset within each chunk
    const int   col = ln & 7;              // clamped B column (always in range)
    const float msk = (ln < 8) ? 1.f : 0.f;

    v8f acc = {0.f, 0.f, 0.f, 0.f, 0.f, 0.f, 0.f, 0.f};
#pragma unroll
    for (int c = 0; c < 4; ++c) {
        const int j0 = 4 * c + k0;
        v2f A, B;
        A.x = fmaxf(fmaf(h2, Wl1[j0],     bl1[j0]),     0.f);   // v1[j0]
        A.y = fmaxf(fmaf(h2, Wl1[j0 + 1], bl1[j0 + 1]), 0.f);   // v1[j0+1]
        B.x = Wl2[j0 * 8 + col]       * msk;   // Wl2 [16,8] row-major, cols 8..15 = 0
        B.y = Wl2[(j0 + 1) * 8 + col] * msk;
        acc = __builtin_amdgcn_wmma_f32_16x16x4_f32(
            /*neg_a=*/false, A, /*neg_b=*/false, B,
            /*c_mod=*/(short)0, acc, /*reuse_a=*/false, /*reuse_b=*/false);
    }

    const float bl2n = bl2[col] * msk;
    const float w3n  = Wl3[col] * msk;
    const float bl3v = bl3[0];
#pragma unroll
    for (int r = 0; r < 8; ++r) {
        float v = fmaxf(acc[r] + bl2n, 0.f) * w3n;   // relu(v2) * Wl3[n]
        v += __shfl_xor(v, 1, 32);                   // sum over n=0..7 (lanes 0-7
        v += __shfl_xor(v, 2, 32);                   //  and 16-23 independently)
        v += __shfl_xor(v, 4, 32);
        if (ln == 0) y[base + r + hi * 8] = v + bl3v;
    }
}

// ---------------------------------------------------------------- pool
__global__ void k_pool(const float* __restrict__ y, const int* __restrict__ batch,
                       float* __restrict__ gsum, unsigned* __restrict__ gcnt) {
    __shared__ float    ls[N_GRAPHS];
    __shared__ unsigned lc[N_GRAPHS];
    if (threadIdx.x < N_GRAPHS) { ls[threadIdx.x] = 0.f; lc[threadIdx.x] = 0u; }
    __syncthreads();
    for (int n = blockIdx.x * blockDim.x + threadIdx.x; n < N_NODES;
         n += gridDim.x * blockDim.x) {
        int g = batch[n];
        atomicAdd(&ls[g], y[n]);   // ds_add_f32
        atomicAdd(&lc[g], 1u);
    }
    __syncthreads();
    if (threadIdx.x < N_GRAPHS) {
        gatomic_add(&gsum[threadIdx.x], ls[threadIdx.x]);
        atomicAdd(&gcnt[threadIdx.x], lc[threadIdx.x]);
    }
}

__global__ void k_out(const float* __restrict__ gsum, const unsigned* __restrict__ gcnt,
                      float* __restrict__ out) {
    int g = threadIdx.x;
    if (g < N_GRAPHS) out[g] = gsum[g] / fmaxf((float)gcnt[g], 1.0f);
}

// ----------------------------------------------------------------
extern "C" void kernel_launch(void* const* d_in, const int* in_sizes, int n_in,
                              void* d_out, int out_size, void* d_ws, size_t ws_size,
                              hipStream_t stream) {
    const float* x     = (const float*)d_in[0];
    const int*   ei    = (const int*)d_in[1];      // [2, E]: src row then dst row
    const int*   src   = ei;
    const int*   dst   = ei + N_EDGES;
    const int*   batch = (const int*)d_in[2];
    const float* W1  = (const float*)d_in[3];
    const float* b1  = (const float*)d_in[4];
    const float* W2  = (const float*)d_in[5];
    const float* b2  = (const float*)d_in[6];
    const float* Wl1 = (const float*)d_in[7];
    const float* bl1 = (const float*)d_in[8];
    const float* Wl2 = (const float*)d_in[9];
    const float* bl2 = (const float*)d_in[10];
    const float* Wl3 = (const float*)d_in[11];
    const float* bl3 = (const float*)d_in[12];

    char* w = (char*)d_ws;
    unsigned* deg  = (unsigned*)w;  w += (size_t)N_NODES * 4;
    float*    dinv = (float*)w;     w += (size_t)N_NODES * 4;
    float*    p    = (float*)w;     w += (size_t)N_NODES * 4;
    float*    agg1 = (float*)w;     w += (size_t)N_NODES * 4;
    float*    q    = (float*)w;     w += (size_t)N_NODES * 4;
    float*    agg2 = (float*)w;     w += (size_t)N_NODES * 4;
    float*    yv   = (float*)w;     w += (size_t)N_NODES * 4;
    float*    gsum = (float*)w;     w += (size_t)N_GRAPHS * 4;
    unsigned* gcnt = (unsigned*)w;
    float*    out  = (float*)d_out;

    const int EB = N_EDGES / (4 * 256);   // 8192 blocks, 4 edges/lane
    const int NB = N_NODES / 256;         // 512 blocks

    k_init        <<<NB, 256, 0, stream>>>(deg, gsum, gcnt);
    k_degree      <<<EB, 256, 0, stream>>>(dst, deg);
    k_node1       <<<NB, 256, 0, stream>>>(x, deg, dinv, p, agg1);
    k_edge_scatter<<<EB, 256, 0, stream>>>(src, dst, p, agg1);
    k_node2       <<<NB, 256, 0, stream>>>(dinv, agg1, W1, b1, W2, q, agg2);
    k_edge_scatter<<<EB, 256, 0, stream>>>(src, dst, q, agg2);
    k_mlp         <<<N_NODES / 128, 256, 0, stream>>>(dinv, agg2, b2, Wl1, bl1,
                                                      Wl2, bl2, Wl3, bl3, yv);
    k_pool        <<<512, 256, 0, stream>>>(yv, batch, gsum, gcnt);
    k_out         <<<1, 64, 0, stream>>>(gsum, gcnt, out);
}